// BoneRefusion_28604482191880
// MI455X (gfx1250) — compile-verified
//
#include <hip/hip_runtime.h>

typedef __attribute__((ext_vector_type(2))) float v2f;
typedef __attribute__((ext_vector_type(8))) float v8f;

#define NGROUPS 17
#define WAVES_PER_BLOCK 8
#define TILE 16            // tokens per WMMA tile
#define TPW  (2 * TILE)    // tokens per wave (two interleaved tiles)
#define XROW 48            // floats per token row (16 limbs * 3 ch)
#define OROW 51            // floats per token output row (17 groups * 3 ch)

__global__ __launch_bounds__(WAVES_PER_BLOCK * 32)
void bone_mlp_wmma_kernel(const float* __restrict__ x,
                          const float* __restrict__ W1,
                          const float* __restrict__ b1,
                          const float* __restrict__ W2,
                          const float* __restrict__ b2,
                          const int*   __restrict__ idx,
                          float* __restrict__ out,
                          int npairs)
{
    __shared__ float lds_out[WAVES_PER_BLOCK][TPW * OROW]; // 8 * 6528B = ~52KB

    const int lane = threadIdx.x & 31;
    const int wave = threadIdx.x >> 5;
    const int pair = blockIdx.x * WAVES_PER_BLOCK + wave;   // two tiles per wave
    if (pair >= npairs) return;             // wave-uniform: EXEC stays all-1s

    const int half = lane >> 4;             // 0: lanes 0-15, 1: lanes 16-31
    const int lm   = lane & 15;
    const int cidx = lm < 3 ? lm : 2;       // clamped output-channel index

    // Lane lm gathers token (pair*32 + lm) for tile A, (+16) for tile B.
    const float* xrowA = x + (size_t)(pair * TPW + lm) * XROW;
    const float* xrowB = xrowA + TILE * XROW;
    float* lrow = &lds_out[wave][0];

    for (int g = 0; g < NGROUPS; ++g) {
        const float* w1g  = W1 + g * 12 * 16;        // W1[g][k][h]
        const float* w2g  = W2 + g * 16 * 3 + cidx;  // W2[g][h][cidx]
        const int*   idxg = idx + g * 4;

        // ---------------- Layer 1:  Hᵀ = W1ᵀ (16x12) × Gᵀ (12x16), two tiles ------
        // A chunk j, VGPR v: K_hw = 4j + v + 2*half, value = W1[g][K][h=lm] (shared)
        // B chunk j, VGPR v: K_hw = 4j + v + 2*half, value = g_feat[token=lm][K]
        v8f accA = {};
        v8f accB = {};
#pragma unroll
        for (int j = 0; j < 3; ++j) {
            const int f0 = 4 * j + 2 * half;
            const int f1 = f0 + 1;
            const int o0 = idxg[f0 / 3] * 3 + f0 % 3;
            const int o1 = idxg[f1 / 3] * 3 + f1 % 3;
            v2f a, bA, bB;
            a.x  = w1g[f0 * 16 + lm];
            a.y  = w1g[f1 * 16 + lm];
            bA.x = xrowA[o0];  bA.y = xrowA[o1];
            bB.x = xrowB[o0];  bB.y = xrowB[o1];
            accA = __builtin_amdgcn_wmma_f32_16x16x4_f32(
                false, a, false, bA, (short)0, accA, false, false);
            accB = __builtin_amdgcn_wmma_f32_16x16x4_f32(
                false, a, false, bB, (short)0, accB, false, false);
        }

        // acc VGPR i: lanes0-15 -> h[token lm][hidden i], lanes16-31 -> hidden i+8
        const float* b1g = b1 + g * 16 + 8 * half;
#pragma unroll
        for (int i = 0; i < 8; ++i) {
            const float bb = b1g[i];
            float tA = accA[i] + bb;
            float tB = accB[i] + bb;
            accA[i] = tA > 0.0f ? tA : 0.0f;  // bias + ReLU, per-lane
            accB[i] = tB > 0.0f ? tB : 0.0f;
        }

        // ---------------- Layer 2:  Out = H (16x16) × W2 (16x3, N padded) ---------
        // Feed acc VGPR pairs (2j, 2j+1) as A chunk j. Hardware K order for chunk j
        // maps to hidden {2j, 2j+1, 2j+8, 2j+9}; permute W2 rows to match:
        // B chunk j, VGPR v: hidden = 2j + v + 8*half, N=c. Output columns c>=3 are
        // never stored, so the W2 column index is clamped, not masked.
        v8f acc2A = {};
        v8f acc2B = {};
#pragma unroll
        for (int j = 0; j < 4; ++j) {
            const int h0 = 2 * j + 8 * half;
            v2f b;
            b.x = w2g[(h0 + 0) * 3];
            b.y = w2g[(h0 + 1) * 3];
            v2f aA; aA.x = accA[2 * j]; aA.y = accA[2 * j + 1];
            v2f aB; aB.x = accB[2 * j]; aB.y = accB[2 * j + 1];
            acc2A = __builtin_amdgcn_wmma_f32_16x16x4_f32(
                false, aA, false, b, (short)0, acc2A, false, false);
            acc2B = __builtin_amdgcn_wmma_f32_16x16x4_f32(
                false, aB, false, b, (short)0, acc2B, false, false);
        }

        // acc2 VGPR i: token = i + 8*half (within tile), c = lm (valid for lm < 3)
        const float bb2 = b2[g * 3 + cidx];   // lanes lm>=3 never store, clamp is fine
        if (lm < 3) {
#pragma unroll
            for (int i = 0; i < 8; ++i) {
                const int row = i + 8 * half;
                lrow[row * OROW + g * 3 + lm]                = acc2A[i] + bb2;
                lrow[(row + TILE) * OROW + g * 3 + lm]       = acc2B[i] + bb2;
            }
        }
    }

    // Coalesced non-temporal writeback: 1632 contiguous floats per wave
    // (= 51 full-width iterations, no tail). Same wave wrote this LDS slice;
    // DS ops are in-order within a wave; no barrier needed.
    float* outp = out + (size_t)pair * (TPW * OROW);
#pragma unroll 4
    for (int k = lane; k < TPW * OROW; k += 32)
        __builtin_nontemporal_store(lrow[k], &outp[k]);
}

extern "C" void kernel_launch(void* const* d_in, const int* in_sizes, int n_in,
                              void* d_out, int out_size, void* d_ws, size_t ws_size,
                              hipStream_t stream) {
    (void)n_in; (void)out_size; (void)d_ws; (void)ws_size;
    const float* x  = (const float*)d_in[0];
    const float* W1 = (const float*)d_in[1];
    const float* b1 = (const float*)d_in[2];
    const float* W2 = (const float*)d_in[3];
    const float* b2 = (const float*)d_in[4];
    const int*   idx = (const int*)d_in[5];
    float* out = (float*)d_out;

    const int ntok   = in_sizes[0] / XROW;          // 497664
    const int npairs = ntok / TPW;                  // 15552 (exact)
    const int blocks = (npairs + WAVES_PER_BLOCK - 1) / WAVES_PER_BLOCK;

    bone_mlp_wmma_kernel<<<blocks, WAVES_PER_BLOCK * 32, 0, stream>>>(
        x, W1, b1, W2, b2, idx, out, npairs);
}